// Downsampling_33311766347847
// MI455X (gfx1250) — compile-verified
//
#include <hip/hip_runtime.h>

// CDNA5 / gfx1250: wave32, WMMA. Full-fp32 DFT-as-GEMM pipeline using
// V_WMMA_F32_16X16X4_F32 (exact fp32, matches the f32 FFT reference).
// v2: 32x128 register blocking per wave (2 M-tiles) -> every B fragment
// feeds 2 WMMAs, halving VMEM issue and L2 traffic per FLOP.

typedef __attribute__((ext_vector_type(2))) float vf2;
typedef __attribute__((ext_vector_type(8))) float vf8;

#define NWIN   1022
#define HOP    256
#define T_IN   127
#define BATCH  64
#define T_OUT  62
#define M1     8128      // 64*127 inverse-DFT rows
#define M2     3968      // 64*62 forward-DFT rows
#define M1DT   254       // 8128/32 double-tiles
#define M2DT   124       // 3968/32 double-tiles
#define SIGP   16896     // padded keep-signal length (66*256)
#define LKEEP  16670

// d_ws float-offset layout (~46 MB total)
#define OFF_TINV   0u
#define OFF_TFWD   1048576u                 // 512 pairs * 1024 cols * 2
#define OFF_FRAMES 2097152u
#define OFF_SIG    10420224u                // 2097152 + 8128*1024

// ---------------- table construction ----------------

__device__ __forceinline__ float hann_w(int i) {
  return (float)(0.5 - 0.5 * cos(6.283185307179586 * (double)i / 1022.0));
}

__device__ __forceinline__ void sincos1022(int k, int n, float* s, float* c) {
  // reduce k*n mod 1022 in integers so the fp argument stays in [0, 2pi)
  int mm = (int)(((long long)k * (long long)n) % 1022ll);
  double th = 6.283185307179586 * (double)mm / 1022.0;
  *s = (float)sin(th);
  *c = (float)cos(th);
}

// T_inv pair layout: T[p*1024 + n] = {B[2p][n], B[2p+1][n]}, p in [0,512)
// rows 0..511  : Re coefs  c_k * cos(2pi k n / N) / N * SYNTH[n]  (c=1 for k=0,511 else 2)
// rows 512..1023: Im coefs -2 * sin(2pi k n / N) / N * SYNTH[n]
__global__ void init_tinv_kernel(vf2* __restrict__ T) {
  int id = blockIdx.x * blockDim.x + threadIdx.x;
  if (id >= 512 * 1024) return;
  int p = id >> 10;
  int n = id & 1023;
  vf2 v; v.x = 0.f; v.y = 0.f;
  if (n < NWIN) {
    // synth window = w / d,  d[j] = sum_m w^2[m*256 + j]
    float w = hann_w(n);
    int j = n & 255;
    float d = 0.f;
    for (int q = 0; q < 4; ++q) {
      int i = q * 256 + j;
      if (i < NWIN) { float wi = hann_w(i); d += wi * wi; }
    }
    float scale = (w / d) / 1022.f;
    int r0 = 2 * p;
    float s0, c0, s1, c1;
    if (r0 < 512) {
      int k0 = r0, k1 = r0 + 1;
      sincos1022(k0, n, &s0, &c0);
      sincos1022(k1, n, &s1, &c1);
      float f0 = (k0 == 0 || k0 == 511) ? 1.f : 2.f;
      float f1 = (k1 == 0 || k1 == 511) ? 1.f : 2.f;
      v.x = f0 * c0 * scale;
      v.y = f1 * c1 * scale;
    } else {
      int k0 = r0 - 512, k1 = r0 - 511;
      sincos1022(k0, n, &s0, &c0);
      sincos1022(k1, n, &s1, &c1);
      v.x = -2.f * s0 * scale;   // zero naturally at k=0,511
      v.y = -2.f * s1 * scale;
    }
  }
  T[id] = v;
}

// T_fwd pair layout: T[p*1024 + col] = {B[2p][col], B[2p+1][col]}, p = n/2
// col 0..511 : Re(S)[k] coefs  cos(2pi k n / N) * WINDOW[n]
// col 512..  : Im(S)[k] coefs -sin(2pi k n / N) * WINDOW[n]
// rows n=1022,1023 are zero padding.
__global__ void init_tfwd_kernel(vf2* __restrict__ T) {
  int id = blockIdx.x * blockDim.x + threadIdx.x;
  if (id >= 512 * 1024) return;
  int p = id >> 10;
  int col = id & 1023;
  int k = col & 511;
  bool isim = (col >= 512);
  vf2 v; v.x = 0.f; v.y = 0.f;
  int n0 = 2 * p, n1 = 2 * p + 1;
  if (n0 < NWIN) {
    float s, c; sincos1022(k, n0, &s, &c);
    v.x = (isim ? -s : c) * hann_w(n0);
  }
  if (n1 < NWIN) {
    float s, c; sincos1022(k, n1, &s, &c);
    v.y = (isim ? -s : c) * hann_w(n1);
  }
  T[id] = v;
}

// ---------------- GEMM 1: inverse DFT (+synth) ----------------
// frames(8128 x 1022) = [Re|Im](8128 x 1024) @ T_inv(1024 x 1022)
// A read directly from the (64,512,127,2) input; row m = b*127 + t.
// Each wave: 32 rows x 128 cols (2 M-tiles share every B fragment).
__global__ __launch_bounds__(256) void gemm_inv_kernel(
    const float* __restrict__ in, const vf2* __restrict__ T,
    float* __restrict__ frames)
{
  const int wave  = threadIdx.x >> 5;
  const int lane  = threadIdx.x & 31;
  const int dmt   = blockIdx.x * 8 + wave;   // double M-tile
  if (dmt >= M1DT) return;                   // uniform per wave: EXEC stays all-1
  const int ngrp = blockIdx.y;               // 0..7 -> 128 output cols
  const int ln = lane & 15;
  const int hi = lane >> 4;
  const int m0 = dmt * 32 + ln;              // A row, tile 0
  const int m1 = m0 + 16;                    // A row, tile 1
  const int b0 = m0 / 127, t0 = m0 - b0 * 127;
  const int b1 = m1 / 127, t1 = m1 - b1 * 127;
  // element kk<512: Re bin kk -> in[((b*512+kk)*127+t)*2]
  // element kk>=512: Im bin kk-512 -> ... + 1
  const float* arow0 = in + (size_t)b0 * 130048 + (size_t)t0 * 2;
  const float* arow1 = in + (size_t)b1 * 130048 + (size_t)t1 * 2;
  vf8 acc0[8] = {};
  vf8 acc1[8] = {};

  // K steps 0..127: Re bins (kk = 4ks + 2hi in [0,510])
  for (int ks = 0; ks < 128; ++ks) {
    const int kk = ks * 4 + 2 * hi;
    vf2 a0, a1;
    a0.x = arow0[kk * 254];       a0.y = arow0[kk * 254 + 254];
    a1.x = arow1[kk * 254];       a1.y = arow1[kk * 254 + 254];
    const vf2* brow = T + (size_t)(ks * 2 + hi) * 1024 + ngrp * 128 + ln;
#pragma unroll
    for (int jj = 0; jj < 8; ++jj) {
      vf2 bf = brow[jj * 16];
      acc0[jj] = __builtin_amdgcn_wmma_f32_16x16x4_f32(
          false, a0, false, bf, (short)0, acc0[jj], false, false);
      acc1[jj] = __builtin_amdgcn_wmma_f32_16x16x4_f32(
          false, a1, false, bf, (short)0, acc1[jj], false, false);
    }
  }
  // K steps 128..255: Im bins
  for (int ks = 128; ks < 256; ++ks) {
    const int kk = ks * 4 + 2 * hi - 512;
    vf2 a0, a1;
    a0.x = arow0[kk * 254 + 1];   a0.y = arow0[kk * 254 + 255];
    a1.x = arow1[kk * 254 + 1];   a1.y = arow1[kk * 254 + 255];
    const vf2* brow = T + (size_t)(ks * 2 + hi) * 1024 + ngrp * 128 + ln;
#pragma unroll
    for (int jj = 0; jj < 8; ++jj) {
      vf2 bf = brow[jj * 16];
      acc0[jj] = __builtin_amdgcn_wmma_f32_16x16x4_f32(
          false, a0, false, bf, (short)0, acc0[jj], false, false);
      acc1[jj] = __builtin_amdgcn_wmma_f32_16x16x4_f32(
          false, a1, false, bf, (short)0, acc1[jj], false, false);
    }
  }

  // C/D layout: lane l element v -> row = v + 8*(l>>4), col = l&15
#pragma unroll
  for (int jj = 0; jj < 8; ++jj) {
    float* col = frames + ngrp * 128 + jj * 16 + ln;
#pragma unroll
    for (int v = 0; v < 8; ++v) {
      const int r = v + 8 * hi;
      col[(size_t)(dmt * 32 + r) * 1024] = acc0[jj][v];
      col[(size_t)(dmt * 32 + 16 + r) * 1024] = acc1[jj][v];
    }
  }
}

// ---------------- overlap-add + decimate (gather, deterministic) ----------------
__global__ void ola_kernel(const float* __restrict__ frames, float* __restrict__ sig) {
  int idx = blockIdx.x * blockDim.x + threadIdx.x;
  if (idx >= BATCH * SIGP) return;
  const int b = idx / SIGP;
  const int s = idx - b * SIGP;
  float acc = 0.f;
  if (s < LKEEP) {
    const int p = (s < 16608) ? (s << 1) : (s + 16608);   // KEEP mapping
    const int tmin = (p >= NWIN) ? ((p - 766) >> 8) : 0;  // ceil((p-1021)/256)
    int tmax = p >> 8; if (tmax > T_IN - 1) tmax = T_IN - 1;
    const float* fb = frames + (size_t)b * T_IN * 1024;
    for (int t = tmin; t <= tmax; ++t)
      acc += fb[(size_t)t * 1024 + (p - (t << 8))];
  }
  sig[idx] = acc;
}

// ---------------- GEMM 2: forward DFT (+window), epilogue -> output layout ----
// S(3968 x 1024) = fr(3968 x 1022) @ T_fwd(1022 x 1024); fr gathered in-place
// from sig (frame start t2*256, always 8B aligned). Output scattered to
// (64, 512, 62, 2). Each wave: 32 rows x 128 cols.
__global__ __launch_bounds__(256) void gemm_fwd_kernel(
    const float* __restrict__ sig, const vf2* __restrict__ T,
    float* __restrict__ out)
{
  const int wave  = threadIdx.x >> 5;
  const int lane  = threadIdx.x & 31;
  const int dmt   = blockIdx.x * 8 + wave;   // double M-tile, 0..123
  if (dmt >= M2DT) return;                   // uniform per wave
  const int ngrp = blockIdx.y;
  const int ln = lane & 15;
  const int hi = lane >> 4;
  const int m0 = dmt * 32 + ln;
  const int m1 = m0 + 16;
  const int b0 = m0 / T_OUT, t20 = m0 - b0 * T_OUT;
  const int b1 = m1 / T_OUT, t21 = m1 - b1 * T_OUT;
  const vf2* ap0 = (const vf2*)(sig + (size_t)b0 * SIGP + (size_t)t20 * 256);
  const vf2* ap1 = (const vf2*)(sig + (size_t)b1 * SIGP + (size_t)t21 * 256);
  vf8 acc0[8] = {};
  vf8 acc1[8] = {};

  for (int ks = 0; ks < 256; ++ks) {
    vf2 a0 = ap0[ks * 2 + hi];               // samples k0+2hi, k0+2hi+1
    vf2 a1 = ap1[ks * 2 + hi];
    const vf2* brow = T + (size_t)(ks * 2 + hi) * 1024 + ngrp * 128 + ln;
#pragma unroll
    for (int jj = 0; jj < 8; ++jj) {
      vf2 bf = brow[jj * 16];
      acc0[jj] = __builtin_amdgcn_wmma_f32_16x16x4_f32(
          false, a0, false, bf, (short)0, acc0[jj], false, false);
      acc1[jj] = __builtin_amdgcn_wmma_f32_16x16x4_f32(
          false, a1, false, bf, (short)0, acc1[jj], false, false);
    }
  }

#pragma unroll
  for (int jj = 0; jj < 8; ++jj) {
    const int col = ngrp * 128 + jj * 16 + ln;
    const int cc = col & 511;
    const int ri = col >> 9;                 // 0 = Re, 1 = Im
#pragma unroll
    for (int v = 0; v < 8; ++v) {
      const int r = v + 8 * hi;
      {
        const int mr = dmt * 32 + r;
        const int b2 = mr / T_OUT;
        const int tt = mr - b2 * T_OUT;
        out[(((size_t)b2 * 512 + cc) * T_OUT + tt) * 2 + ri] = acc0[jj][v];
      }
      {
        const int mr = dmt * 32 + 16 + r;
        const int b2 = mr / T_OUT;
        const int tt = mr - b2 * T_OUT;
        out[(((size_t)b2 * 512 + cc) * T_OUT + tt) * 2 + ri] = acc1[jj][v];
      }
    }
  }
}

// ---------------- launcher ----------------
extern "C" void kernel_launch(void* const* d_in, const int* in_sizes, int n_in,
                              void* d_out, int out_size, void* d_ws, size_t ws_size,
                              hipStream_t stream) {
  (void)in_sizes; (void)n_in; (void)out_size; (void)ws_size;
  const float* in = (const float*)d_in[0];
  float* out = (float*)d_out;
  float* ws = (float*)d_ws;

  init_tinv_kernel<<<2048, 256, 0, stream>>>((vf2*)(ws + OFF_TINV));
  init_tfwd_kernel<<<2048, 256, 0, stream>>>((vf2*)(ws + OFF_TFWD));

  gemm_inv_kernel<<<dim3(32, 8), 256, 0, stream>>>(
      in, (const vf2*)(ws + OFF_TINV), ws + OFF_FRAMES);

  ola_kernel<<<(BATCH * SIGP + 255) / 256, 256, 0, stream>>>(
      ws + OFF_FRAMES, ws + OFF_SIG);

  gemm_fwd_kernel<<<dim3(16, 8), 256, 0, stream>>>(
      ws + OFF_SIG, (const vf2*)(ws + OFF_TFWD), out);
}